// InterpretableMultiHeadAttention_34479997452968
// MI455X (gfx1250) — compile-verified
//
#include <hip/hip_runtime.h>

// MI455X / gfx1250, wave32. fp32 WMMA path (V_WMMA_F32_16X16X4_F32).
typedef __attribute__((ext_vector_type(2))) float v2f;
typedef __attribute__((ext_vector_type(4))) float v4f;
typedef __attribute__((ext_vector_type(8))) float v8f;

#define Bc 2
#define Tt 2048
#define Ee 1024
#define Hh 16
#define Dd 64
#define BM 16
#define NT (Tt / 16)  // 128 column tiles per row strip
#define SPAD 4        // strip pad: keeps rows 16B-aligned (2052 % 4 == 0) and
                      // rotates banks by 4*M for conflict-free column reads

// ---------------------------------------------------------------------------
// Kernel 1: per (b, h, 16-row tile): S = QK^T/8, softmax, write attn,
//           head_tile = attn @ V  -> workspace head buffer [B*T, E]
// gridDim = (T/16, B*H), block = 256 threads (8 waves)
// ---------------------------------------------------------------------------
__global__ __launch_bounds__(256) void attn_softmax_pv_kernel(
    const float* __restrict__ Q, const float* __restrict__ K,
    const float* __restrict__ V, float* __restrict__ attnOut,
    float* __restrict__ headOut) {
  const int mblk = blockIdx.x;  // 0..127 row tile within T
  const int bh = blockIdx.y;    // 0..31
  const int b = bh / Hh;
  const int h = bh % Hh;
  const int tid = threadIdx.x;
  const int lane = tid & 31;
  const int wave = tid >> 5;
  const int nl = lane & 15;  // N (and M for A-operand) index, 0..15
  const int hf = lane >> 4;  // lane half select, 0..1

  __shared__ __align__(16) float sQ[BM][Dd + 1];
  __shared__ __align__(16) float sS[BM][Tt + SPAD];  // 16 x 2052 strip, 128KB
  __shared__ __align__(16) float sPart[8][16][16];

  const size_t bRow = (size_t)b * Tt;
  const int row0 = mblk * BM;

  // ---- Phase 1: cooperative coalesced load of Q tile [16 x 64] ----
  {
    const float* qbase = Q + (bRow + row0) * (size_t)Ee + h * Dd;
    for (int i = tid; i < BM * Dd; i += 256) {
      int r = i >> 6, c = i & 63;
      sQ[r][c] = qbase[(size_t)r * Ee + c];
    }
  }
  __syncthreads();

  // ---- Phase 2: S = (Q K^T) * 1/sqrt(64); each wave owns 16 column tiles ---
  for (int n = wave; n < NT; n += 8) {
    const float* kbase = K + (bRow + n * 16 + nl) * (size_t)Ee + h * Dd;
    v8f c0 = {}, c1 = {};
#pragma unroll
    for (int kk = 0; kk < Dd; kk += 8) {
      v2f a, bv;
      // A 16x4 f32 layout: lanes 0-15 hold K=0,1; lanes 16-31 hold K=2,3
      a.x = sQ[nl][kk + 2 * hf];
      a.y = sQ[nl][kk + 2 * hf + 1];
      // B 4x16 f32 layout: VGPR0 rows K={0,1} by lane half, VGPR1 rows K={2,3}
      bv.x = kbase[kk + hf];
      bv.y = kbase[kk + 2 + hf];
      c0 = __builtin_amdgcn_wmma_f32_16x16x4_f32(false, a, false, bv, (short)0,
                                                 c0, false, false);
      a.x = sQ[nl][kk + 4 + 2 * hf];
      a.y = sQ[nl][kk + 4 + 2 * hf + 1];
      bv.x = kbase[kk + 4 + hf];
      bv.y = kbase[kk + 6 + hf];
      c1 = __builtin_amdgcn_wmma_f32_16x16x4_f32(false, a, false, bv, (short)0,
                                                 c1, false, false);
    }
    v8f c = c0 + c1;
#pragma unroll
    for (int r = 0; r < 8; ++r)
      sS[r + 8 * hf][n * 16 + nl] = c[r] * 0.125f;  // 1/sqrt(d_k)
  }
  __syncthreads();

  // ---- Phase 3: row softmax over the 16 x 2048 strip (vectorized b128) ----
  {
    const int srow = tid >> 4;  // 16 threads per row (one half-wave)
    const int scol = tid & 15;  // each owns 4 consecutive cols per step
    v4f* srowp = (v4f*)&sS[srow][0];  // row stride 2052 floats -> 16B aligned
    float mx = -3.402823466e38f;
    for (int j = 0; j < Tt / 64; ++j) {
      v4f x = srowp[scol + 16 * j];
      mx = fmaxf(fmaxf(fmaxf(mx, x.x), x.y), fmaxf(x.z, x.w));
    }
    for (int off = 8; off; off >>= 1) mx = fmaxf(mx, __shfl_xor(mx, off, 16));
    float sum = 0.f;
    for (int j = 0; j < Tt / 64; ++j) {
      v4f x = srowp[scol + 16 * j];
      x.x = __expf(x.x - mx);
      x.y = __expf(x.y - mx);
      x.z = __expf(x.z - mx);
      x.w = __expf(x.w - mx);
      srowp[scol + 16 * j] = x;
      sum += (x.x + x.y) + (x.z + x.w);
    }
    for (int off = 8; off; off >>= 1) sum += __shfl_xor(sum, off, 16);
    const float inv = 1.0f / sum;
    // lanes 0-15 of a half-wave cover 256B contiguous -> global_store_b128
    v4f* arow4 =
        (v4f*)(attnOut + ((size_t)bh * Tt + (row0 + srow)) * (size_t)Tt);
    for (int j = 0; j < Tt / 64; ++j) {
      v4f x = srowp[scol + 16 * j] * inv;
      srowp[scol + 16 * j] = x;  // keep normalized probs for PV
      arow4[scol + 16 * j] = x;  // 537MB output stream, b128, written once
    }
  }
  __syncthreads();

  // ---- Phase 4: head = attn @ V ; wave w -> col tile (w&3), K-half (w>>2) --
  {
    const int nd = wave & 3;   // which 16-col slice of the 64-wide head
    const int kh = wave >> 2;  // which half of K = 2048
    const float* vbase = V + bRow * (size_t)Ee + h * Dd + nd * 16 + nl;
    v8f c0 = {}, c1 = {};
#pragma unroll 2
    for (int ks = 0; ks < Tt / 2; ks += 8) {
      const int kk = kh * (Tt / 2) + ks;
      v2f a, bv;
      a.x = sS[nl][kk + 2 * hf];
      a.y = sS[nl][kk + 2 * hf + 1];
      bv.x = vbase[(size_t)(kk + hf) * Ee];
      bv.y = vbase[(size_t)(kk + 2 + hf) * Ee];
      c0 = __builtin_amdgcn_wmma_f32_16x16x4_f32(false, a, false, bv, (short)0,
                                                 c0, false, false);
      a.x = sS[nl][kk + 4 + 2 * hf];
      a.y = sS[nl][kk + 4 + 2 * hf + 1];
      bv.x = vbase[(size_t)(kk + 4 + hf) * Ee];
      bv.y = vbase[(size_t)(kk + 6 + hf) * Ee];
      c1 = __builtin_amdgcn_wmma_f32_16x16x4_f32(false, a, false, bv, (short)0,
                                                 c1, false, false);
    }
    v8f c = c0 + c1;
#pragma unroll
    for (int r = 0; r < 8; ++r) sPart[wave][r + 8 * hf][nl] = c[r];
  }
  __syncthreads();

  // combine the two K-halves and write the 16x64 head tile
  for (int i = tid; i < 16 * Dd; i += 256) {
    int r = i >> 6, cc = i & 63;
    int nd2 = cc >> 4, c2 = cc & 15;
    float v = sPart[nd2][r][c2] + sPart[nd2 + 4][r][c2];
    headOut[(bRow + row0 + r) * (size_t)Ee + h * Dd + cc] = v;
  }
}

// ---------------------------------------------------------------------------
// Kernel 2: out = head @ W_out^T.  M=B*T=4096, N=E=1024, K=E=1024 (f32 WMMA).
// WG tile 64 rows x 128 cols; wave w owns one 16-wide column strip, 4 row
// tiles. gridDim = (4096/64, 1024/128), block = 256.
// ---------------------------------------------------------------------------
__global__ __launch_bounds__(256) void out_proj_kernel(
    const float* __restrict__ Hd, const float* __restrict__ W,
    float* __restrict__ Out) {
  const int rtile = blockIdx.x;
  const int ctile = blockIdx.y;
  const int tid = threadIdx.x;
  const int lane = tid & 31;
  const int wave = tid >> 5;
  const int nl = lane & 15;
  const int hf = lane >> 4;

  const int col0 = ctile * 128 + wave * 16;
  // W_out is [out,in] row-major => B[k][n] = W[n][k]
  const float* wrow = W + (size_t)(col0 + nl) * Ee;

  for (int rt = 0; rt < 4; ++rt) {
    const int row0 = rtile * 64 + rt * 16;
    const float* arow = Hd + (size_t)(row0 + nl) * Ee;
    v8f c0 = {}, c1 = {};
#pragma unroll 2
    for (int kk = 0; kk < Ee; kk += 8) {
      v2f a, bv;
      a.x = arow[kk + 2 * hf];
      a.y = arow[kk + 2 * hf + 1];
      bv.x = wrow[kk + hf];
      bv.y = wrow[kk + 2 + hf];
      c0 = __builtin_amdgcn_wmma_f32_16x16x4_f32(false, a, false, bv, (short)0,
                                                 c0, false, false);
      a.x = arow[kk + 4 + 2 * hf];
      a.y = arow[kk + 4 + 2 * hf + 1];
      bv.x = wrow[kk + 4 + hf];
      bv.y = wrow[kk + 6 + hf];
      c1 = __builtin_amdgcn_wmma_f32_16x16x4_f32(false, a, false, bv, (short)0,
                                                 c1, false, false);
    }
    v8f c = c0 + c1;
#pragma unroll
    for (int r = 0; r < 8; ++r)
      Out[(size_t)(row0 + r + 8 * hf) * Ee + col0 + nl] = c[r];
  }
}

extern "C" void kernel_launch(void* const* d_in, const int* in_sizes, int n_in,
                              void* d_out, int out_size, void* d_ws,
                              size_t ws_size, hipStream_t stream) {
  (void)in_sizes;
  (void)n_in;
  (void)out_size;
  (void)ws_size;
  const float* Q = (const float*)d_in[0];
  const float* K = (const float*)d_in[1];
  const float* V = (const float*)d_in[2];
  const float* W = (const float*)d_in[3];

  float* out = (float*)d_out;                // [B*T, E]
  float* attn = out + (size_t)Bc * Tt * Ee;  // [B*H, T, T]
  float* head = (float*)d_ws;                // [B*T, E] scratch (16MB)

  dim3 g1(Tt / BM, Bc * Hh);
  attn_softmax_pv_kernel<<<g1, 256, 0, stream>>>(Q, K, V, attn, head);

  dim3 g2((Bc * Tt) / 64, Ee / 128);
  out_proj_kernel<<<g2, 256, 0, stream>>>(head, W, out);
}